// LSTM_72095321030674
// MI455X (gfx1250) — compile-verified
//
#include <hip/hip_runtime.h>
#include <hip/hip_bf16.h>

typedef __attribute__((ext_vector_type(2))) float v2f;
typedef __attribute__((ext_vector_type(8))) float v8f;

#define T_STEPS 8192
#define IN_DIM  3000
#define H_DIM   100
#define G_DIM   400   // 4*H

// ---------------------------------------------------------------------------
// Kernel 1: x_gates[t][n] = sum_k input_seq[t][k] * w_ih[n][k] + b_ih[n] + b_hh[n]
// M=8192, N=400, K=3000 GEMM using V_WMMA_F32_16X16X4_F32.
// Each wave computes a 16(M) x 80(N) strip: 5 accumulators reuse one A frag.
// Block = 128 threads = 4 waves stacked in M -> block covers 64(M) x 80(N).
// Grid = (400/80, 8192/64) = (5, 128).
// ---------------------------------------------------------------------------
__global__ __launch_bounds__(128) void lstm_gates_gemm(
    const float* __restrict__ A,     // input_seq [8192][3000]
    const float* __restrict__ W,     // w_ih      [400][3000]
    const float* __restrict__ b_ih,  // [400]
    const float* __restrict__ b_hh,  // [400]
    float* __restrict__ xg)          // out: [8192][400]
{
    const int lane = threadIdx.x & 31;
    const int wave = threadIdx.x >> 5;
    const int m0 = (blockIdx.y * 4 + wave) * 16;  // row tile base
    const int n0 = blockIdx.x * 80;               // col strip base (5 x 16)

    // A fragment (16x4 f32): lanes 0-15 -> rows m, K=0,1 ; lanes 16-31 -> K=2,3
    const int mr = lane & 15;
    const int kb = (lane >> 4) << 1;              // 0 or 2
    const float* arow  = A + (size_t)(m0 + mr) * IN_DIM + kb;
    // B fragment (4x16 f32): lane holds column n = lane&15, same K split.
    // B[k][n] = w_ih[n][k] (transpose is free: read along w_ih row).
    const float* wrow0 = W + (size_t)(n0 + mr) * IN_DIM + kb;

    v8f acc0 = {}, acc1 = {}, acc2 = {}, acc3 = {}, acc4 = {};

    for (int k = 0; k < IN_DIM; k += 4) {
        v2f a = *(const v2f*)(arow + k);
        v2f b0 = *(const v2f*)(wrow0 + (size_t)0 * 16 * IN_DIM + k);
        v2f b1 = *(const v2f*)(wrow0 + (size_t)1 * 16 * IN_DIM + k);
        v2f b2 = *(const v2f*)(wrow0 + (size_t)2 * 16 * IN_DIM + k);
        v2f b3 = *(const v2f*)(wrow0 + (size_t)3 * 16 * IN_DIM + k);
        v2f b4 = *(const v2f*)(wrow0 + (size_t)4 * 16 * IN_DIM + k);
        acc0 = __builtin_amdgcn_wmma_f32_16x16x4_f32(false, a, false, b0, (short)0, acc0, false, false);
        acc1 = __builtin_amdgcn_wmma_f32_16x16x4_f32(false, a, false, b1, (short)0, acc1, false, false);
        acc2 = __builtin_amdgcn_wmma_f32_16x16x4_f32(false, a, false, b2, (short)0, acc2, false, false);
        acc3 = __builtin_amdgcn_wmma_f32_16x16x4_f32(false, a, false, b3, (short)0, acc3, false, false);
        acc4 = __builtin_amdgcn_wmma_f32_16x16x4_f32(false, a, false, b4, (short)0, acc4, false, false);
    }

    // C/D layout: VGPR r -> lanes 0-15: (M=r, N=lane) ; lanes 16-31: (M=8+r, N=lane-16)
    const int nc = lane & 15;
    const int mbase = m0 + ((lane >> 4) << 3);
    v8f accs[5] = {acc0, acc1, acc2, acc3, acc4};
#pragma unroll
    for (int t = 0; t < 5; ++t) {
        const int n = n0 + t * 16 + nc;
        const float bias = b_ih[n] + b_hh[n];
#pragma unroll
        for (int r = 0; r < 8; ++r) {
            xg[(size_t)(mbase + r) * G_DIM + n] = accs[t][r] + bias;
        }
    }
}

// ---------------------------------------------------------------------------
// Kernel 2: persistent single-workgroup LSTM scan. 400 threads; thread j owns
// gate row j with its 100 w_hh weights held in registers. h broadcast via LDS.
// Final 100-dot projection + b_lin done by thread 0 at the end.
// ---------------------------------------------------------------------------
__device__ __forceinline__ float sigm_f(float x) {
    return 1.0f / (1.0f + __expf(-x));
}
__device__ __forceinline__ float tanh_f(float x) {
    return 1.0f - 2.0f / (__expf(2.0f * x) + 1.0f);
}

__global__ __launch_bounds__(400, 1) void lstm_scan_kernel(
    const float* __restrict__ xg,     // [8192][400] precomputed gates
    const float* __restrict__ w_hh,   // [400][100]
    const float* __restrict__ w_lin,  // [100]
    const float* __restrict__ b_lin,  // [1]
    float* __restrict__ out)          // [1]
{
    __shared__ float h_lds[H_DIM];
    __shared__ float g_lds[G_DIM];

    const int j = threadIdx.x;  // 0..399, all valid

    // Pin this gate-row of w_hh in registers (100 VGPRs/lane).
    float wr[H_DIM];
#pragma unroll
    for (int k = 0; k < H_DIM; ++k) wr[k] = w_hh[(size_t)j * H_DIM + k];

    float c = 0.0f;
    if (j < H_DIM) h_lds[j] = 0.0f;
    __syncthreads();

    const float* xgp = xg + j;
    for (int t = 0; t < T_STEPS; ++t) {
        // Independent of h: issue global load early.
        float xv = xgp[(size_t)t * G_DIM];

        // g_j = xv + sum_k h[k] * w_hh[j][k], 4 partial sums to pipeline FMAs.
        float4 s = make_float4(0.f, 0.f, 0.f, 0.f);
#pragma unroll
        for (int k4 = 0; k4 < H_DIM / 4; ++k4) {
            float4 hv = *(const float4*)(&h_lds[k4 * 4]);
            s.x = fmaf(hv.x, wr[4 * k4 + 0], s.x);
            s.y = fmaf(hv.y, wr[4 * k4 + 1], s.y);
            s.z = fmaf(hv.z, wr[4 * k4 + 2], s.z);
            s.w = fmaf(hv.w, wr[4 * k4 + 3], s.w);
        }
        g_lds[j] = xv + (s.x + s.y) + (s.z + s.w);
        __syncthreads();

        if (j < H_DIM) {
            const float ig = sigm_f(g_lds[j]);
            const float fg = sigm_f(g_lds[j + H_DIM]);
            const float gg = tanh_f(g_lds[j + 2 * H_DIM]);
            const float og = sigm_f(g_lds[j + 3 * H_DIM]);
            c = fg * c + ig * gg;
            h_lds[j] = og * tanh_f(c);
        }
        __syncthreads();
    }

    if (j == 0) {
        float acc = b_lin[0];
        for (int k = 0; k < H_DIM; ++k) acc = fmaf(h_lds[k], w_lin[k], acc);
        out[0] = acc;
    }
}

// ---------------------------------------------------------------------------
extern "C" void kernel_launch(void* const* d_in, const int* in_sizes, int n_in,
                              void* d_out, int out_size, void* d_ws, size_t ws_size,
                              hipStream_t stream) {
    const float* input_seq = (const float*)d_in[0];  // [8192*3000]
    const float* w_ih      = (const float*)d_in[1];  // [400*3000]
    const float* w_hh      = (const float*)d_in[2];  // [400*100]
    const float* b_ih      = (const float*)d_in[3];  // [400]
    const float* b_hh      = (const float*)d_in[4];  // [400]
    const float* w_lin     = (const float*)d_in[5];  // [100]
    const float* b_lin     = (const float*)d_in[6];  // [1]
    float* out = (float*)d_out;
    float* xg  = (float*)d_ws;                       // [8192*400] fp32 scratch

    dim3 grid1(5, 128);   // N strips of 80, M blocks of 64
    lstm_gates_gemm<<<grid1, 128, 0, stream>>>(input_seq, w_ih, b_ih, b_hh, xg);

    lstm_scan_kernel<<<1, 400, 0, stream>>>(xg, w_hh, w_lin, b_lin, out);
}